// Tisa_27066883899429
// MI455X (gfx1250) — compile-verified
//
#include <hip/hip_runtime.h>

#define SEQ 2048
#define SLEN (2 * SEQ + 1)   // 4097 score entries per head
#define NH 16
#define NK 5

typedef __attribute__((ext_vector_type(4))) float fvec4;

// ---------------------------------------------------------------------------
// Kernel 1: build scores[h, r] = sum_k amp[k,h] * exp(-|w[k,h]| * d^2),
//           d = off[k,h] - (r - SEQ).   Table: 16 x 4097 floats (262 KB).
// ---------------------------------------------------------------------------
__global__ __launch_bounds__(256) void build_scores(
    const float* __restrict__ off, const float* __restrict__ wid,
    const float* __restrict__ amp, float* __restrict__ scores) {
  const int r = blockIdx.x * blockDim.x + threadIdx.x;
  const int h = blockIdx.y;
  if (r >= SLEN) return;
  const float rel = (float)(r - SEQ);
  float acc = 0.0f;
#pragma unroll
  for (int k = 0; k < NK; ++k) {
    const float d = off[k * NH + h] - rel;
    acc += amp[k * NH + h] * __expf(-fabsf(wid[k * NH + h]) * d * d);
  }
  scores[h * SLEN + r] = acc;
}

// ---------------------------------------------------------------------------
// Kernel 2: one block per (h, i) row. LDS slice lds[x] = scores[h, i + x],
//           x in [0, 2047]; then out[h,i,j] = lds[2047 - j] (row-invariant!).
//           Slice is fetched with CDNA5 async global->LDS loads, and the
//           256 MB output is written with 128-bit non-temporal stores.
// ---------------------------------------------------------------------------
template <bool USE_WS>
__global__ __launch_bounds__(512) void toeplitz_fill(
    const float* __restrict__ scores, const float* __restrict__ off,
    const float* __restrict__ wid, const float* __restrict__ amp,
    float* __restrict__ out) {
  __shared__ __align__(16) float lds[SEQ + 4];
  const int i = blockIdx.x;   // row index
  const int h = blockIdx.y;   // head
  const int t = threadIdx.x;  // 0..511

  if (USE_WS) {
    const float* src = scores + (size_t)h * SLEN + i;  // need elements [0,2047]
#if defined(__has_builtin) && \
    __has_builtin(__builtin_amdgcn_global_load_async_to_lds_b32) && \
    __has_builtin(__builtin_amdgcn_s_wait_asynccnt)
    // CDNA5 async path: global_load_async_to_lds_b32 (ASYNCcnt-tracked)
#pragma unroll
    for (int e = 0; e < 4; ++e) {
      __builtin_amdgcn_global_load_async_to_lds_b32(
          (__attribute__((address_space(1))) int*)(
              (const __attribute__((address_space(1))) float*)src + t + e * 512),
          (__attribute__((address_space(3))) int*)(
              (__attribute__((address_space(3))) float*)&lds[0] + t + e * 512),
          0, 0);
    }
    __builtin_amdgcn_s_wait_asynccnt(0);
    __syncthreads();
#else
#pragma unroll
    for (int e = 0; e < 4; ++e) lds[t + e * 512] = src[t + e * 512];
    __syncthreads();
#endif
  } else {
    // ws too small: recompute the slice in-block (5 exps per element)
    float o[NK], w[NK], a[NK];
#pragma unroll
    for (int k = 0; k < NK; ++k) {
      o[k] = off[k * NH + h];
      w[k] = fabsf(wid[k * NH + h]);
      a[k] = amp[k * NH + h];
    }
#pragma unroll
    for (int e = 0; e < 4; ++e) {
      const int x = t + e * 512;
      const float rel = (float)(i + x - SEQ);
      float acc = 0.0f;
#pragma unroll
      for (int k = 0; k < NK; ++k) {
        const float d = o[k] - rel;
        acc += a[k] * __expf(-w[k] * d * d);
      }
      lds[x] = acc;
    }
    __syncthreads();
  }

  // out[h, i, 4t .. 4t+3] = lds[2047-4t], lds[2046-4t], lds[2045-4t], lds[2044-4t]
  // -> one aligned ds_load_b128 + register reverse + NT global_store_b128
  const fvec4 q = *(const fvec4*)&lds[2044 - 4 * t];
  fvec4 v;
  v.x = q.w; v.y = q.z; v.z = q.y; v.w = q.x;
  fvec4* dst = (fvec4*)(out + ((size_t)(h * SEQ + i)) * SEQ) + t;
  __builtin_nontemporal_store(v, dst);
}

// ---------------------------------------------------------------------------
// Host launcher
// ---------------------------------------------------------------------------
extern "C" void kernel_launch(void* const* d_in, const int* in_sizes, int n_in,
                              void* d_out, int out_size, void* d_ws,
                              size_t ws_size, hipStream_t stream) {
  // setup_inputs order: seq_len (scalar), kernel_offsets, kernel_widths,
  //                     kernel_amplitudes  (each [K=5, H=16] float32)
  const float* off = (const float*)d_in[1];
  const float* wid = (const float*)d_in[2];
  const float* amp = (const float*)d_in[3];
  float* out = (float*)d_out;

  const size_t need = (size_t)NH * SLEN * sizeof(float);  // 262 KB score table
  if (ws_size >= need) {
    float* scores = (float*)d_ws;
    build_scores<<<dim3((SLEN + 255) / 256, NH), 256, 0, stream>>>(off, wid,
                                                                   amp, scores);
    toeplitz_fill<true><<<dim3(SEQ, NH), 512, 0, stream>>>(scores, off, wid,
                                                           amp, out);
  } else {
    toeplitz_fill<false><<<dim3(SEQ, NH), 512, 0, stream>>>(nullptr, off, wid,
                                                            amp, out);
  }
}